// GCN_71184787964325
// MI455X (gfx1250) — compile-verified
//
#include <hip/hip_runtime.h>
#include <hip/hip_bf16.h>

typedef float v2f __attribute__((ext_vector_type(2)));
typedef float v8f __attribute__((ext_vector_type(8)));

#ifndef __HIP_DEVICE_COMPILE__
#endif

// ---------------------------------------------------------------------------
// Degree accumulation: deg[dst[e]] += ew[e]  (hardware FP32 atomics)
// ---------------------------------------------------------------------------
__global__ __launch_bounds__(256) void deg_kernel(const long long* __restrict__ dst,
                                                  const float* __restrict__ ew,
                                                  float* __restrict__ deg,
                                                  long long E) {
  long long e = (long long)blockIdx.x * blockDim.x + threadIdx.x;
  if (e < E) {
    unsafeAtomicAdd(&deg[dst[e]], ew[e]);
  }
}

// deg -> dinv in place: dinv = rsqrt(deg + 1)
__global__ __launch_bounds__(256) void rsqrt_kernel(float* __restrict__ d, long long N) {
  long long i = (long long)blockIdx.x * blockDim.x + threadIdx.x;
  if (i < N) {
    d[i] = __frsqrt_rn(d[i] + 1.0f);
  }
}

// ---------------------------------------------------------------------------
// H[N,128] = X[N,128] @ W[128,128] via V_WMMA_F32_16X16X4_F32.
// Block = 256 threads = 8 waves; block covers 16 rows, wave w covers cols
// [16w, 16w+16).  K loop: 128 / 4 = 32 WMMA issues, fully unrolled.
//
// A tile (16x4 f32, 2 VGPR):  lanes 0-15 -> M=lane, K = {k,k+1};
//                             lanes 16-31 -> M=lane-16, K = {k+2,k+3}.
// B tile (4x16 f32, 2 VGPR):  lanes 0-15 -> N=lane, rows K={k,k+1};
//                             lanes 16-31 -> N=lane-16, rows K={k+2,k+3}.
// C/D (16x16 f32, 8 VGPR):    vgpr v: lanes 0-15 -> M=v; lanes 16-31 -> M=v+8.
// ---------------------------------------------------------------------------
__global__ __launch_bounds__(256) void gemm_x128_wmma(const float* __restrict__ X,
                                                      const float* __restrict__ W,
                                                      float* __restrict__ H,
                                                      long long N) {
  const int wave = threadIdx.x >> 5;          // 0..7 -> column tile
  const int lane = threadIdx.x & 31;
  const long long row0 = (long long)blockIdx.x * 16;
  if (row0 >= N) return;

  const int m    = lane & 15;
  const int half = lane >> 4;                  // 0 or 1
  const int koff = half * 2;                   // K sub-offset: 0 or 2
  const int col  = wave * 16 + m;

  long long arow = row0 + m;
  if (arow >= N) arow = N - 1;                 // clamp (N is a multiple of 16 here)
  const float* __restrict__ xrow = X + arow * 128;

  v8f c = {};
#pragma unroll
  for (int kk = 0; kk < 128; kk += 4) {
    v2f a, b;
    a.x = xrow[kk + koff];
    a.y = xrow[kk + koff + 1];
    b.x = W[(long long)(kk + koff) * 128 + col];       // coalesced across lanes
    b.y = W[(long long)(kk + koff + 1) * 128 + col];
    c = __builtin_amdgcn_wmma_f32_16x16x4_f32(
        /*neg_a=*/false, a, /*neg_b=*/false, b,
        /*c_mod=*/(short)0, c, /*reuse_a=*/false, /*reuse_b=*/false);
  }

  const long long rbase = row0 + (half << 3);  // lanes 16-31 hold M=8..15
#pragma unroll
  for (int v = 0; v < 8; ++v) {
    long long r = rbase + v;
    if (r < N) H[r * 128 + col] = c[v];
  }
}

// ---------------------------------------------------------------------------
// Edge scatter: agg[dst] += h[src] * (dinv[src]*ew*dinv[dst]).
// 32 threads per edge, float4 per thread -> 4 hardware FP32 atomics.
// h and agg (51.2 MB each) are L2-resident on MI455X (192 MB L2).
// ---------------------------------------------------------------------------
__global__ __launch_bounds__(256) void scatter_agg(const float* __restrict__ H,
                                                   const long long* __restrict__ src,
                                                   const long long* __restrict__ dst,
                                                   const float* __restrict__ ew,
                                                   const float* __restrict__ dinv,
                                                   float* __restrict__ agg,
                                                   long long E) {
  long long i = (long long)blockIdx.x * blockDim.x + threadIdx.x;
  if (i >= E * 32) return;
  const long long e = i >> 5;
  const int q = (int)(i & 31);

  const long long s = src[e];
  const long long t = dst[e];
  const float nrm = dinv[s] * ew[e] * dinv[t];

  const float4 hv = ((const float4*)(H + s * 128))[q];
  float* ap = agg + t * 128 + (long long)q * 4;
  unsafeAtomicAdd(ap + 0, hv.x * nrm);
  unsafeAtomicAdd(ap + 1, hv.y * nrm);
  unsafeAtomicAdd(ap + 2, hv.z * nrm);
  unsafeAtomicAdd(ap + 3, hv.w * nrm);
}

// ---------------------------------------------------------------------------
// out[n,d] = relu(agg[n,d] + h[n,d]*dinv[n]^2 + bias[d])
// ---------------------------------------------------------------------------
__global__ __launch_bounds__(256) void finalize_relu(const float* __restrict__ agg,
                                                     const float* __restrict__ h,
                                                     const float* __restrict__ dinv,
                                                     const float* __restrict__ bias,
                                                     float* __restrict__ out,
                                                     long long total) {
  long long i = (long long)blockIdx.x * blockDim.x + threadIdx.x;
  if (i >= total) return;
  const long long n = i >> 7;
  const int d = (int)(i & 127);
  const float di = dinv[n];
  const float v = agg[i] + h[i] * di * di + bias[d];
  out[i] = v > 0.0f ? v : 0.0f;
}

// ---------------------------------------------------------------------------
// out[n] = X[n,:] . Wl + bl   (wave per node, shuffle reduction, wave32)
// ---------------------------------------------------------------------------
__global__ __launch_bounds__(256) void linear_out(const float* __restrict__ X,
                                                  const float* __restrict__ Wl,
                                                  const float* __restrict__ bl,
                                                  float* __restrict__ out,
                                                  long long N) {
  const int wavesPerBlock = blockDim.x >> 5;
  const long long node = (long long)blockIdx.x * wavesPerBlock + (threadIdx.x >> 5);
  const int lane = threadIdx.x & 31;
  if (node >= N) return;

  const float4 xv = ((const float4*)(X + node * 128))[lane];
  const float4 wv = ((const float4*)Wl)[lane];
  float s = xv.x * wv.x + xv.y * wv.y + xv.z * wv.z + xv.w * wv.w;
#pragma unroll
  for (int off = 16; off > 0; off >>= 1) s += __shfl_down(s, off, 32);
  if (lane == 0) out[node] = s + bl[0];
}

// ---------------------------------------------------------------------------
extern "C" void kernel_launch(void* const* d_in, const int* in_sizes, int n_in,
                              void* d_out, int out_size, void* d_ws, size_t ws_size,
                              hipStream_t stream) {
  const float* x        = (const float*)d_in[0];
  const long long* eidx = (const long long*)d_in[1];   // int64 per reference
  const float* ew       = (const float*)d_in[2];
  const float* W1       = (const float*)d_in[3];
  const float* b1       = (const float*)d_in[4];
  const float* W2       = (const float*)d_in[5];
  const float* b2       = (const float*)d_in[6];
  const float* Wl       = (const float*)d_in[7];
  const float* bl       = (const float*)d_in[8];

  const long long N = (long long)in_sizes[0] / 128;
  const long long E = (long long)in_sizes[2];
  const long long* src = eidx;
  const long long* dst = eidx + E;

  // workspace layout: dinv[N] | buf0[N*128] | buf1[N*128]
  char* ws = (char*)d_ws;
  float* dinv = (float*)ws;
  size_t off = ((size_t)N * sizeof(float) + 511) & ~(size_t)511;
  const size_t bsz = (size_t)N * 128 * sizeof(float);
  float* buf0 = (float*)(ws + off);
  float* buf1 = (float*)(ws + off + bsz);

  const int T = 256;
  const unsigned gE   = (unsigned)((E + T - 1) / T);
  const unsigned gN   = (unsigned)((N + T - 1) / T);
  const unsigned gNE  = (unsigned)((N * 128 + T - 1) / T);
  const unsigned gSC  = (unsigned)((E * 32 + T - 1) / T);
  const unsigned gGEMM = (unsigned)((N + 15) / 16);

  // --- degree / dinv ---
  hipMemsetAsync(dinv, 0, (size_t)N * sizeof(float), stream);
  deg_kernel<<<gE, T, 0, stream>>>(dst, ew, dinv, E);
  rsqrt_kernel<<<gN, T, 0, stream>>>(dinv, N);

  // --- layer 1 ---
  gemm_x128_wmma<<<gGEMM, T, 0, stream>>>(x, W1, buf0, N);         // h1 -> buf0
  hipMemsetAsync(buf1, 0, bsz, stream);                             // agg1 -> buf1
  scatter_agg<<<gSC, T, 0, stream>>>(buf0, src, dst, ew, dinv, buf1, E);
  finalize_relu<<<gNE, T, 0, stream>>>(buf1, buf0, dinv, b1, buf0, N * 128); // x1 -> buf0

  // --- layer 2 ---
  gemm_x128_wmma<<<gGEMM, T, 0, stream>>>(buf0, W2, buf1, N);      // h2 -> buf1
  hipMemsetAsync(buf0, 0, bsz, stream);                             // agg2 -> buf0
  scatter_agg<<<gSC, T, 0, stream>>>(buf1, src, dst, ew, dinv, buf0, E);
  finalize_relu<<<gNE, T, 0, stream>>>(buf0, buf1, dinv, b2, buf0, N * 128); // x2 -> buf0

  // --- readout ---
  linear_out<<<(unsigned)((N + 7) / 8), T, 0, stream>>>(buf0, Wl, bl, (float*)d_out, N);
}